// PointNet2Seg_15496242004856
// MI455X (gfx1250) — compile-verified
//
#include <hip/hip_runtime.h>
#include <hip/hip_bf16.h>

// ---------------------------------------------------------------------------
// PointNet++ MSG segmentation forward, MI455X (gfx1250), wave32.
// Dense MLP layers run on V_WMMA_F32_16X16X4_F32 (exact fp32 tensor path),
// weights staged in LDS (bank-conflict-free), 2 m-tiles per wave.
// ---------------------------------------------------------------------------

typedef float v2f __attribute__((ext_vector_type(2)));
typedef float v8f __attribute__((ext_vector_type(8)));

#define INV_STD 0.99999500003749993f   // 1/sqrt(1+1e-5)

// ------------------------- FPS (farthest point sampling) -------------------
// One workgroup per batch. dist[] lives in LDS (N<=8192 -> 32KB).
// Matches reference: far=0; record far; dist=min(dist,|x-c|^2); far=argmax.
__global__ __launch_bounds__(256) void fps_kernel(
    const float* __restrict__ xyz, int Npts, int npoint, int* __restrict__ out_idx)
{
    __shared__ float dist[8192];
    __shared__ float rv[256];
    __shared__ int   ri[256];
    const int b = blockIdx.x;
    const int t = threadIdx.x;
    const float* p = xyz + (size_t)b * Npts * 3;

    for (int i = t; i < Npts; i += 256) dist[i] = 1e10f;
    __syncthreads();

    int far = 0;
    for (int s = 0; s < npoint; ++s) {
        if (t == 0) out_idx[b * npoint + s] = far;
        const float cx = p[far*3+0], cy = p[far*3+1], cz = p[far*3+2];
        float best = -1.0f; int bi = 0x7fffffff;
        for (int i = t; i < Npts; i += 256) {
            const float dx = p[i*3+0]-cx, dy = p[i*3+1]-cy, dz = p[i*3+2]-cz;
            float d = dx*dx + dy*dy + dz*dz;
            float dm = dist[i];
            dm = d < dm ? d : dm;
            dist[i] = dm;
            if (dm > best) { best = dm; bi = i; }   // per-thread indices ascend
        }
        rv[t] = best; ri[t] = bi;
        __syncthreads();
        for (int off = 128; off > 0; off >>= 1) {
            if (t < off) {
                const float ov = rv[t+off]; const int oi = ri[t+off];
                if (ov > rv[t] || (ov == rv[t] && oi < ri[t])) { rv[t]=ov; ri[t]=oi; }
            }
            __syncthreads();
        }
        far = ri[0];
        __syncthreads();
    }
}

// ------------------------- gather 3D points by index -----------------------
__global__ void gather_points3(const float* __restrict__ pts, const int* __restrict__ idx,
                               float* __restrict__ out, int Btot, int Nsrc, int M)
{
    const int gid = blockIdx.x * blockDim.x + threadIdx.x;
    if (gid >= Btot * M) return;
    const int b = gid / M;
    int j = idx[gid]; if (j >= Nsrc) j = Nsrc - 1;
    const float* s = pts + ((size_t)b * Nsrc + j) * 3;
    out[gid*3+0] = s[0]; out[gid*3+1] = s[1]; out[gid*3+2] = s[2];
}

// ------------------------- ball query (one wave32 per query) ---------------
// Ordered first-K-by-index selection within radius, pad with first hit:
// exact semantics of sort(where(sq>r2, N, arange))[:K] + pad-with-first.
__global__ void ball_query(const float* __restrict__ xyz, const float* __restrict__ new_xyz,
                           int* __restrict__ idx, int Btot, int Npts, int M, int K, float r2)
{
    const int wid  = (blockIdx.x * blockDim.x + threadIdx.x) >> 5;
    const int lane = threadIdx.x & 31;
    if (wid >= Btot * M) return;           // uniform per wave
    const int b = wid / M;
    const float cx = new_xyz[wid*3+0], cy = new_xyz[wid*3+1], cz = new_xyz[wid*3+2];
    const float* p = xyz + (size_t)b * Npts * 3;

    int count = 0, first = 0; bool haveFirst = false;
    for (int base = 0; base < Npts && count < K; base += 32) {
        const int n = base + lane;
        bool ok = false;
        if (n < Npts) {
            const float dx = p[n*3+0]-cx, dy = p[n*3+1]-cy, dz = p[n*3+2]-cz;
            ok = (dx*dx + dy*dy + dz*dz) <= r2;
        }
        const unsigned mask = (unsigned)__ballot(ok);
        if (ok) {
            const int pos = count + __popc(mask & ((1u << lane) - 1u));
            if (pos < K) idx[(size_t)wid * K + pos] = n;
        }
        if (!haveFirst && mask) { first = base + (__ffs((int)mask) - 1); haveFirst = true; }
        count += __popc(mask);
    }
    const int total = count < K ? count : K;
    for (int pos = total + lane; pos < K; pos += 32)
        idx[(size_t)wid * K + pos] = first;
}

// ------------------------- grouping kernels --------------------------------
__global__ void group_xyz_k(const float* __restrict__ xyz, const float* __restrict__ new_xyz,
                            const int* __restrict__ idx, float* __restrict__ out,
                            int Btot, int Npts, int M, int K)
{
    const int gid = blockIdx.x * blockDim.x + threadIdx.x;
    if (gid >= Btot * M * K) return;
    const int bm = gid / K, b = bm / M;
    int j = idx[gid]; if (j >= Npts) j = Npts - 1;
    const float* s = xyz + ((size_t)b * Npts + j) * 3;
    const float* c = new_xyz + (size_t)bm * 3;
    float* o = out + (size_t)gid * 3;
    o[0] = s[0]-c[0]; o[1] = s[1]-c[1]; o[2] = s[2]-c[2];
}

__global__ void group_feat_k(const float* __restrict__ feat, const float* __restrict__ xyz,
                             const float* __restrict__ new_xyz, const int* __restrict__ idx,
                             float* __restrict__ out, int Btot, int Npts, int M, int K, int C)
{
    const int gid = blockIdx.x * blockDim.x + threadIdx.x;
    if (gid >= Btot * M * K) return;
    const int bm = gid / K, b = bm / M;
    int j = idx[gid]; if (j >= Npts) j = Npts - 1;
    const float* f = feat + ((size_t)b * Npts + j) * C;
    float* o = out + (size_t)gid * (C + 3);
    for (int c = 0; c < C; ++c) o[c] = f[c];
    const float* s = xyz + ((size_t)b * Npts + j) * 3;
    const float* q = new_xyz + (size_t)bm * 3;
    o[C+0] = s[0]-q[0]; o[C+1] = s[1]-q[1]; o[C+2] = s[2]-q[2];
}

// ------------------------- fused MLP layer on fp32 WMMA --------------------
// out = relu(g * ((A@W + b) * INV_STD) + bt).
// Block = 4 waves; block owns one 16-col slab of W, K-chunked into LDS.
// Each wave computes TWO 16x16 m-tiles, so each ds_load_b64 B-fragment
// feeds two v_wmma_f32_16x16x4_f32. LDS layout [col][k], KPAD=KT+4 =>
// lane half0 hits banks {4c,4c+1}, half1 {4c+2,4c+3}: conflict-free b64.
// Fill path: each thread moves a 4-wide k-group per column -> 4 clause-able
// global_load_b32 + one aligned ds_store_b128 (guards only on last group).
// A fragment (16x4 f32, ISA 7.12.2): lanes 0-15 row=lane K={k0,k0+1};
// lanes 16-31 row=lane-16 K={k0+2,k0+3}. B mirrored across lane halves.
#define KT   256
#define KPAD (KT + 4)

__global__ __launch_bounds__(128) void mlp_gemm_wmma(
    const float* __restrict__ A, const float* __restrict__ W,
    const float* __restrict__ bias, const float* __restrict__ gamma,
    const float* __restrict__ beta, float* __restrict__ Out,
    int R, int Cin, int Cout, int ostride, int coloff)
{
    __shared__ float Wlds[16 * KPAD];
    const int lane = threadIdx.x;                 // 0..31
    const int wv   = threadIdx.y;                 // 0..3
    const int tid  = wv * 32 + lane;
    const int half = lane >> 4, l16 = lane & 15;

    const int tileA = blockIdx.x * 8 + wv * 2;    // two m-tiles per wave
    const int row0a = tileA * 16, row0b = row0a + 16;
    const int col0  = blockIdx.y * 16;
    const int bc    = col0 + l16;                 // < Cout by construction

    int ra = row0a + l16; if (ra >= R) ra = R - 1;   // clamp; W zero-pad kills tail
    int rb = row0b + l16; if (rb >= R) rb = R - 1;
    const float* __restrict__ Aa = A + (size_t)ra * Cin;
    const float* __restrict__ Ab = A + (size_t)rb * Cin;

    v8f acc0 = {}, acc1 = {};
    for (int k0 = 0; k0 < Cin; k0 += KT) {
        const int kend  = (Cin - k0 < KT) ? (Cin - k0) : KT;
        const int kpad4 = (kend + 3) & ~3;        // uniform across block
        const int ngrp  = kpad4 >> 2;

        __syncthreads();                          // protect previous chunk reads
        for (int i = tid; i < 16 * ngrp; i += 128) {
            const int c  = i & 15;
            const int k4 = (i >> 4) * 4;
            const float* __restrict__ wp = W + (size_t)(k0 + k4) * Cout + col0 + c;
            float4 v;
            if (k4 + 4 <= kend) {                 // full group: 4 clause-able loads
                v.x = wp[0];
                v.y = wp[(size_t)Cout];
                v.z = wp[(size_t)Cout * 2];
                v.w = wp[(size_t)Cout * 3];
            } else {                              // last (partial) group only
                v.x = (k4 + 0 < kend) ? wp[0]                 : 0.0f;
                v.y = (k4 + 1 < kend) ? wp[(size_t)Cout]      : 0.0f;
                v.z = (k4 + 2 < kend) ? wp[(size_t)Cout * 2]  : 0.0f;
                v.w = (k4 + 3 < kend) ? wp[(size_t)Cout * 3]  : 0.0f;
            }
            *(float4*)&Wlds[c * KPAD + k4] = v;   // 16B-aligned ds_store_b128
        }
        __syncthreads();
        __builtin_prefetch(Aa + k0 + KT, 0, 1);   // next A chunk -> global_prefetch_b8
        __builtin_prefetch(Ab + k0 + KT, 0, 1);

        const int kmain = kend & ~3;
        const float* __restrict__ bld = &Wlds[l16 * KPAD + 2 * half];

        #pragma unroll 2
        for (int kk = 0; kk < kmain; kk += 4) {
            const int ac = k0 + kk + 2 * half;    // in-bounds: ac+1 < Cin here
            v2f a0; a0.x = Aa[ac]; a0.y = Aa[ac + 1];
            v2f a1; a1.x = Ab[ac]; a1.y = Ab[ac + 1];
            const v2f bf = *(const v2f*)(bld + kk);
            acc0 = __builtin_amdgcn_wmma_f32_16x16x4_f32(false, a0, false, bf,
                                                         (short)0, acc0, false, false);
            acc1 = __builtin_amdgcn_wmma_f32_16x16x4_f32(false, a1, false, bf,
                                                         (short)0, acc1, false, false);
        }
        if (kmain < kend) {                       // ragged K tail (Cin = 3/131/259)
            const int ac = k0 + kmain + 2 * half;
            v2f a0, a1;
            a0.x = (ac     < Cin) ? Aa[ac]     : 0.0f;
            a0.y = (ac + 1 < Cin) ? Aa[ac + 1] : 0.0f;
            a1.x = (ac     < Cin) ? Ab[ac]     : 0.0f;
            a1.y = (ac + 1 < Cin) ? Ab[ac + 1] : 0.0f;
            const v2f bf = *(const v2f*)(bld + kmain);   // zero-padded in LDS
            acc0 = __builtin_amdgcn_wmma_f32_16x16x4_f32(false, a0, false, bf,
                                                         (short)0, acc0, false, false);
            acc1 = __builtin_amdgcn_wmma_f32_16x16x4_f32(false, a1, false, bf,
                                                         (short)0, acc1, false, false);
        }
    }

    const float bv = bias[bc], gv = gamma[bc], btv = beta[bc];
    #pragma unroll
    for (int r = 0; r < 8; ++r) {                 // C/D: VGPR r -> row r + 8*half
        const int o0 = row0a + r + half * 8;
        if (o0 < R) {
            const float y = gv * ((acc0[r] + bv) * INV_STD) + btv;
            Out[(size_t)o0 * ostride + coloff + bc] = fmaxf(y, 0.0f);
        }
        const int o1 = row0b + r + half * 8;
        if (o1 < R) {
            const float y = gv * ((acc1[r] + bv) * INV_STD) + btv;
            Out[(size_t)o1 * ostride + coloff + bc] = fmaxf(y, 0.0f);
        }
    }
}

// ------------------------- max over K neighbors ----------------------------
__global__ void maxpool_k(const float* __restrict__ in, float* __restrict__ out,
                          int RowsOut, int K, int C, int ostride, int coloff)
{
    const int gid = blockIdx.x * blockDim.x + threadIdx.x;
    if (gid >= RowsOut * C) return;
    const int r = gid / C, c = gid % C;
    const float* p = in + (size_t)r * K * C + c;
    float m = p[0];
    for (int k = 1; k < K; ++k) { const float v = p[(size_t)k * C]; m = v > m ? v : m; }
    out[(size_t)r * ostride + coloff + c] = m;
}

// ------------------------- SA3 / FP3 concat builders -----------------------
__global__ void sa3_build(const float* __restrict__ l2_p, const float* __restrict__ l2_xyz,
                          float* __restrict__ out, int Rows /*B*128*/)
{
    const int gid = blockIdx.x * blockDim.x + threadIdx.x;
    if (gid >= Rows * 259) return;
    const int row = gid / 259, c = gid % 259;
    out[(size_t)row * 259 + c] = (c < 256) ? l2_p[(size_t)row * 256 + c]
                                           : l2_xyz[(size_t)row * 3 + (c - 256)];
}

__global__ void fp3_build(const float* __restrict__ l2_p, const float* __restrict__ l3_p,
                          float* __restrict__ out, int Rows /*B*128*/)
{
    const int gid = blockIdx.x * blockDim.x + threadIdx.x;
    if (gid >= Rows * 768) return;
    const int row = gid / 768, c = gid % 768, b = row / 128;
    out[(size_t)row * 768 + c] = (c < 256) ? l2_p[(size_t)row * 256 + c]
                                           : l3_p[(size_t)b * 512 + (c - 256)];
}

// ------------------------- 3-NN inverse-distance interpolation -------------
// out row = [feat1 (C1) | interp(feat2) (C2)]. Strict '<' keeps the lower
// index on ties, matching jax.lax.top_k tie-breaking.
__global__ void fp_interp(const float* __restrict__ xyz1, const float* __restrict__ xyz2,
                          const float* __restrict__ feat2, const float* __restrict__ feat1,
                          float* __restrict__ out, int Btot, int M1, int S, int C2, int C1)
{
    const int gid = blockIdx.x * blockDim.x + threadIdx.x;
    if (gid >= Btot * M1) return;
    const int b = gid / M1;
    const float px = xyz1[gid*3+0], py = xyz1[gid*3+1], pz = xyz1[gid*3+2];
    const float* q = xyz2 + (size_t)b * S * 3;

    float d0 = 1e30f, d1 = 1e30f, d2 = 1e30f;
    int   i0 = 0,     i1 = 0,     i2 = 0;
    for (int s = 0; s < S; ++s) {
        const float dx = q[s*3+0]-px, dy = q[s*3+1]-py, dz = q[s*3+2]-pz;
        const float d = dx*dx + dy*dy + dz*dz;
        if (d < d0)      { d2=d1; i2=i1; d1=d0; i1=i0; d0=d; i0=s; }
        else if (d < d1) { d2=d1; i2=i1; d1=d;  i1=s; }
        else if (d < d2) { d2=d;  i2=s; }
    }
    float w0 = 1.0f/(fmaxf(d0,0.0f)+1e-8f);
    float w1 = 1.0f/(fmaxf(d1,0.0f)+1e-8f);
    float w2 = 1.0f/(fmaxf(d2,0.0f)+1e-8f);
    const float ws = w0 + w1 + w2; w0 /= ws; w1 /= ws; w2 /= ws;

    float* o = out + (size_t)gid * (C1 + C2);
    if (feat1) {
        const float* f1 = feat1 + (size_t)gid * C1;
        for (int c = 0; c < C1; ++c) o[c] = f1[c];
    }
    const float* f = feat2 + (size_t)b * S * C2;
    for (int c = 0; c < C2; ++c)
        o[C1 + c] = w0*f[(size_t)i0*C2+c] + w1*f[(size_t)i1*C2+c] + w2*f[(size_t)i2*C2+c];
}

// ------------------------- seg / coord heads (128 -> 3 each) ---------------
__global__ void head_out(const float* __restrict__ h,
                         const float* __restrict__ wseg, const float* __restrict__ bseg,
                         const float* __restrict__ wco,  const float* __restrict__ bco,
                         float* __restrict__ out_seg, float* __restrict__ out_co, int Rows)
{
    const int r = blockIdx.x * blockDim.x + threadIdx.x;
    if (r >= Rows) return;
    const float* x = h + (size_t)r * 128;
    for (int o = 0; o < 3; ++o) {
        float s = bseg[o], c = bco[o];
        for (int i = 0; i < 128; ++i) {
            const float xv = x[i];
            s += xv * wseg[i*3+o];
            c += xv * wco [i*3+o];
        }
        out_seg[(size_t)r*3+o] = s;
        out_co [(size_t)r*3+o] = c;
    }
}

// ===========================================================================
// Host orchestration
// ===========================================================================
struct Layer { const float *w, *b, *g, *bt; int cin, cout; };

extern "C" void kernel_launch(void* const* d_in, const int* in_sizes, int n_in,
                              void* d_out, int out_size, void* d_ws, size_t ws_size,
                              hipStream_t stream) {
    (void)in_sizes; (void)n_in; (void)out_size; (void)ws_size;
    const int B = 8, N = 8192;

    // ---- parse inputs in setup_inputs() insertion order -------------------
    int ip = 0;
    auto nextp = [&]() { return (const float*)d_in[ip++]; };
    auto rdL = [&](int cin, int cout) {
        Layer L; L.w = nextp(); L.b = nextp(); L.g = nextp(); L.bt = nextp();
        L.cin = cin; L.cout = cout; return L;
    };
    const float* xyz = nextp();
    Layer sa1m0[3] = { rdL(3,32),   rdL(32,32),  rdL(32,64)   };
    Layer sa1m1[3] = { rdL(3,32),   rdL(32,32),  rdL(32,64)   };
    Layer sa2m0[3] = { rdL(131,64), rdL(64,64),  rdL(64,128)  };
    Layer sa2m1[3] = { rdL(131,64), rdL(64,64),  rdL(64,128)  };
    Layer sa3m [3] = { rdL(259,128),rdL(128,256),rdL(256,512) };
    Layer fp3L [2] = { rdL(768,256),rdL(256,256) };
    Layer fp2L [2] = { rdL(384,128),rdL(128,128) };
    Layer fp1L [3] = { rdL(128,128),rdL(128,128),rdL(128,128) };
    Layer headL    = rdL(128,128);
    const float* wseg = nextp(); const float* bseg = nextp();
    const float* wco  = nextp(); const float* bco  = nextp();

    // ---- workspace bump allocator -----------------------------------------
    size_t off = 0;
    auto alloc = [&](size_t bytes) -> void* {
        void* p = (char*)d_ws + off;
        off = (off + bytes + 255) & ~(size_t)255;
        return p;
    };
    int*   fps1_idx = (int*)  alloc(sizeof(int)   * B * 512);
    int*   fps2_idx = (int*)  alloc(sizeof(int)   * B * 128);
    int*   bq_idx   = (int*)  alloc(sizeof(int)   * B * 512 * 64);
    float* l1_xyz   = (float*)alloc(sizeof(float) * B * 512 * 3);
    float* l2_xyz   = (float*)alloc(sizeof(float) * B * 128 * 3);
    float* l1_p     = (float*)alloc(sizeof(float) * B * 512 * 128);
    float* l2_p     = (float*)alloc(sizeof(float) * B * 128 * 256);
    float* l3_p     = (float*)alloc(sizeof(float) * B * 512);
    float* l2_up    = (float*)alloc(sizeof(float) * B * 128 * 256);
    float* l1_up    = (float*)alloc(sizeof(float) * B * 512 * 128);
    const size_t BUFE = (size_t)262144 * 64;          // 16.8M floats = 64MB
    float* bufA = (float*)alloc(sizeof(float) * BUFE);
    float* bufB = (float*)alloc(sizeof(float) * BUFE);

    auto nb = [](int n) { return dim3((unsigned)((n + 255) / 256)); };
    auto gemm = [&](const float* A, const Layer& L, int R, float* Out, int ostride, int coloff) {
        const int tiles_m = (R + 15) / 16;
        dim3 grid((unsigned)((tiles_m + 7) / 8), (unsigned)(L.cout / 16));
        mlp_gemm_wmma<<<grid, dim3(32, 4), 0, stream>>>(A, L.w, L.b, L.g, L.bt,
                                                        Out, R, L.cin, L.cout, ostride, coloff);
    };

    // ---- SA1: FPS 8192->512, two radius branches --------------------------
    fps_kernel<<<B, 256, 0, stream>>>(xyz, N, 512, fps1_idx);
    gather_points3<<<nb(B*512), 256, 0, stream>>>(xyz, fps1_idx, l1_xyz, B, N, 512);

    const float r1[2] = {0.1f, 0.2f}; const int k1[2] = {32, 64};
    const Layer* m1[2] = {sa1m0, sa1m1};
    for (int br = 0; br < 2; ++br) {
        const int K = k1[br], R = B * 512 * K;
        ball_query<<<nb(B*512*32), 256, 0, stream>>>(xyz, l1_xyz, bq_idx, B, N, 512, K, r1[br]*r1[br]);
        group_xyz_k<<<nb(R), 256, 0, stream>>>(xyz, l1_xyz, bq_idx, bufA, B, N, 512, K);
        gemm(bufA, m1[br][0], R, bufB, 32, 0);
        gemm(bufB, m1[br][1], R, bufA, 32, 0);
        gemm(bufA, m1[br][2], R, bufB, 64, 0);
        maxpool_k<<<nb(B*512*64), 256, 0, stream>>>(bufB, l1_p, B*512, K, 64, 128, br*64);
    }

    // ---- SA2: FPS 512->128, features 131ch --------------------------------
    fps_kernel<<<B, 256, 0, stream>>>(l1_xyz, 512, 128, fps2_idx);
    gather_points3<<<nb(B*128), 256, 0, stream>>>(l1_xyz, fps2_idx, l2_xyz, B, 512, 128);

    const float r2a[2] = {0.4f, 0.8f}; const int k2[2] = {32, 64};
    const Layer* m2[2] = {sa2m0, sa2m1};
    for (int br = 0; br < 2; ++br) {
        const int K = k2[br], R = B * 128 * K;
        ball_query<<<nb(B*128*32), 256, 0, stream>>>(l1_xyz, l2_xyz, bq_idx, B, 512, 128, K, r2a[br]*r2a[br]);
        group_feat_k<<<nb(R), 256, 0, stream>>>(l1_p, l1_xyz, l2_xyz, bq_idx, bufA, B, 512, 128, K, 128);
        gemm(bufA, m2[br][0], R, bufB, 64, 0);
        gemm(bufB, m2[br][1], R, bufA, 64, 0);
        gemm(bufA, m2[br][2], R, bufB, 128, 0);
        maxpool_k<<<nb(B*128*128), 256, 0, stream>>>(bufB, l2_p, B*128, K, 128, 256, br*128);
    }

    // ---- SA3: global group over the 128 points ----------------------------
    {
        const int R = B * 128;
        sa3_build<<<nb(R*259), 256, 0, stream>>>(l2_p, l2_xyz, bufA, R);
        gemm(bufA, sa3m[0], R, bufB, 128, 0);
        gemm(bufB, sa3m[1], R, bufA, 256, 0);
        gemm(bufA, sa3m[2], R, bufB, 512, 0);
        maxpool_k<<<nb(B*512), 256, 0, stream>>>(bufB, l3_p, B, 128, 512, 512, 0);
    }

    // ---- FP3: broadcast l3 -> l2 ------------------------------------------
    {
        const int R = B * 128;
        fp3_build<<<nb(R*768), 256, 0, stream>>>(l2_p, l3_p, bufA, R);
        gemm(bufA, fp3L[0], R, bufB, 256, 0);
        gemm(bufB, fp3L[1], R, l2_up, 256, 0);
    }

    // ---- FP2: 3-NN interp l2 -> l1 ----------------------------------------
    {
        const int R = B * 512;
        fp_interp<<<nb(R), 256, 0, stream>>>(l1_xyz, l2_xyz, l2_up, l1_p, bufA, B, 512, 128, 256, 128);
        gemm(bufA, fp2L[0], R, bufB, 128, 0);
        gemm(bufB, fp2L[1], R, l1_up, 128, 0);
    }

    // ---- FP1: 3-NN interp l1 -> full cloud, then head + outputs -----------
    {
        const int R = B * N;
        fp_interp<<<nb(R), 256, 0, stream>>>(xyz, l1_xyz, l1_up, nullptr, bufA, B, N, 512, 128, 0);
        gemm(bufA, fp1L[0], R, bufB, 128, 0);
        gemm(bufB, fp1L[1], R, bufA, 128, 0);
        gemm(bufA, fp1L[2], R, bufB, 128, 0);
        gemm(bufB, headL,   R, bufA, 128, 0);   // h in bufA

        float* out_seg = (float*)d_out;
        float* out_co  = out_seg + (size_t)B * N * 3;
        head_out<<<nb(R), 256, 0, stream>>>(bufA, wseg, bseg, wco, bco, out_seg, out_co, R);
    }
}